// TotalLoss_87290915324054
// MI455X (gfx1250) — compile-verified
//
#include <hip/hip_runtime.h>
#include <hip/hip_bf16.h>

typedef float v2f __attribute__((ext_vector_type(2)));
typedef float v8f __attribute__((ext_vector_type(8)));

// ---------------------------------------------------------------------------
// WMMA-based reduction of 256 floats in LDS, executed by one full wave (32
// lanes, EXEC all-1s). Uses V_WMMA_F32_16X16X4_F32 with B = ones(4x16):
//   D[m,n] = sum_k A[m,k] * 1 + C[m,n]
// Four chained WMMAs fold 4 * 64 = 256 values into 16 row-sums (replicated
// across all 16 columns). Each lane then sums its 8 D VGPRs (M = r or r+8
// depending on lane half) and one shfl_xor(16) joins the halves.
// Exact f32 math (RNE), deterministic.
// ---------------------------------------------------------------------------
__device__ __forceinline__ float wmma_reduce256(const float* lds, int lane) {
    v8f c = {};
    v2f b;  b.x = 1.0f; b.y = 1.0f;          // all-ones B, layout-invariant
    const int m     = lane & 15;              // A row for this lane
    const int khalf = (lane >> 4) << 1;       // lanes 16-31 hold K=2,3
#pragma unroll
    for (int i = 0; i < 4; ++i) {
        v2f a;
        a.x = lds[i * 64 + m * 4 + khalf + 0];  // A[m][khalf]   -> VGPR0
        a.y = lds[i * 64 + m * 4 + khalf + 1];  // A[m][khalf+1] -> VGPR1
        c = __builtin_amdgcn_wmma_f32_16x16x4_f32(
                /*neg_a=*/false, a, /*neg_b=*/false, b,
                /*c_mod=*/(short)0, c, /*reuse_a=*/false, /*reuse_b=*/false);
    }
    // Per-lane: sum of 8 M-rows (column value identical for every N).
    float s = c[0] + c[1] + c[2] + c[3] + c[4] + c[5] + c[6] + c[7];
    // Join M=0..7 (lanes 0-15) with M=8..15 (lanes 16-31).
    s += __shfl_xor(s, 16, 32);
    return s;
}

__global__ void loss_zero_kernel(float* acc) {
    if (threadIdx.x < 3) acc[threadIdx.x] = 0.0f;
}

__global__ __launch_bounds__(256)
void loss_main_kernel(const float* __restrict__ resnet,
                      const float* __restrict__ gru,
                      const float* __restrict__ avg_pred,
                      const float* __restrict__ y,
                      const float* __restrict__ avg_hr,
                      float* __restrict__ acc,
                      int n, int m) {
    const int tid    = blockIdx.x * blockDim.x + threadIdx.x;
    const int stride = gridDim.x * blockDim.x;

    float a0 = 0.0f, a1 = 0.0f, a2 = 0.0f;

    // ---- sum |resnet - y| : float4 streaming (global_load_b128) ----
    const int n4 = n >> 2;                    // N divisible by 4
    const float4* r4 = (const float4*)resnet;
    const float4* y4 = (const float4*)y;
    for (int i = tid; i < n4; i += stride) {
        float4 rv = r4[i], yv = y4[i];
        a0 += fabsf(rv.x - yv.x) + fabsf(rv.y - yv.y)
            + fabsf(rv.z - yv.z) + fabsf(rv.w - yv.w);
    }

    // ---- sum |avg_pred - avg_hr| : float4 streaming ----
    const int m4 = m >> 2;                    // M divisible by 4
    const float4* p4 = (const float4*)avg_pred;
    const float4* h4 = (const float4*)avg_hr;
    for (int i = tid; i < m4; i += stride) {
        float4 pv = p4[i], hv = h4[i];
        a1 += fabsf(pv.x - hv.x) + fabsf(pv.y - hv.y)
            + fabsf(pv.z - hv.z) + fabsf(pv.w - hv.w);
    }

    // ---- smoothness: one 6-float chunk per thread, 3x float2 (b64 loads) ----
    const int chunks = n / 6;
    const float2* g2 = (const float2*)gru;    // 24B chunk stride is 8B-aligned
    for (int c = tid; c < chunks; c += stride) {
        float2 x0 = g2[c * 3 + 0];
        float2 x1 = g2[c * 3 + 1];
        float2 x2 = g2[c * 3 + 2];
        float mean = (x0.x + x0.y + x1.x + x1.y + x2.x + x2.y) * (1.0f / 6.0f);
        a2 += fabsf(x0.x - mean) + fabsf(x0.y - mean)
            + fabsf(x1.x - mean) + fabsf(x1.y - mean)
            + fabsf(x2.x - mean) + fabsf(x2.y - mean);
    }

    // ---- block reduction via the matrix unit ----
    __shared__ float lds[3 * 256];
    lds[          threadIdx.x] = a0;
    lds[256 +     threadIdx.x] = a1;
    lds[512 +     threadIdx.x] = a2;
    __syncthreads();

    if (threadIdx.x < 32) {                   // wave 0 only: EXEC all-1s
        const int lane = threadIdx.x;
        float s0 = wmma_reduce256(lds,       lane);
        float s1 = wmma_reduce256(lds + 256, lane);
        float s2 = wmma_reduce256(lds + 512, lane);
        if (lane == 0) {
            atomicAdd(&acc[0], s0);
            atomicAdd(&acc[1], s1);
            atomicAdd(&acc[2], s2);
        }
    }
}

__global__ void loss_final_kernel(const float* __restrict__ acc,
                                  float* __restrict__ out, int n, int m) {
    if (threadIdx.x == 0) {
        float l1     = acc[0] / (float)n + acc[1] / (float)m;
        float smooth = acc[2] * (1.0f / 6.0f);
        float total  = l1 + 100.0f * smooth;
        out[0] = total;
        out[1] = l1;
        out[2] = smooth;
    }
}

extern "C" void kernel_launch(void* const* d_in, const int* in_sizes, int n_in,
                              void* d_out, int out_size, void* d_ws, size_t ws_size,
                              hipStream_t stream) {
    const float* resnet   = (const float*)d_in[0];
    const float* gru      = (const float*)d_in[1];
    const float* avg_pred = (const float*)d_in[2];
    const float* y        = (const float*)d_in[3];
    const float* avg_hr   = (const float*)d_in[4];
    float* out = (float*)d_out;
    float* acc = (float*)d_ws;                // 3 floats of scratch

    const int n = in_sizes[0];                // 3,000,000
    const int m = in_sizes[2];                // 500,000

    const int threads = 256;                  // 8 waves / block (wave32)
    const int blocks  = 2048;                 // grid-stride covers any size

    loss_zero_kernel<<<1, 32, 0, stream>>>(acc);
    loss_main_kernel<<<blocks, threads, 0, stream>>>(resnet, gru, avg_pred, y,
                                                     avg_hr, acc, n, m);
    loss_final_kernel<<<1, 32, 0, stream>>>(acc, out, n, m);
}